// Guide_56212531970411
// MI455X (gfx1250) — compile-verified
//
#include <hip/hip_runtime.h>

// ---------------------------------------------------------------------------
// fe_dep[8,64,256,256] f32, seg[8,21,128,128] f32, w[64,64], b[64].
// Fusion: y = fe@w^T + segup @ (a@w^T) + b, a = (segup.fe)/sum(segup)
// => apply pass is ONE GEMM, K-dim = 64 channels + 32 padded seg classes.
// fp32 tensor path V_WMMA_F32_16X16X4_F32 (confirmed lowering in round 1).
// Async global->LDS (B128, ASYNCcnt) + double-buffered tiles hide HBM latency
// behind the wmma chain and skip the VGPR bounce.
// ---------------------------------------------------------------------------

#define NB   8
#define CCH  64
#define HH   256
#define WW   256
#define HWPX (HH * WW)
#define KCL  21
#define KP   32
#define HS   128
#define JTOT 96     // apply-GEMM K dim: 64 ch + 32 padded classes

#define LPAD 132    // LDS row stride (floats): 128 px + 4 pad, 16B-aligned rows

typedef float v2f __attribute__((ext_vector_type(2)));
typedef float v8f __attribute__((ext_vector_type(8)));
typedef int   v4i_vs __attribute__((vector_size(16)));   // matches builtin param

#if defined(__HIP_DEVICE_COMPILE__) && __has_builtin(__builtin_amdgcn_global_load_async_to_lds_b128)
#define HAVE_ASYNC_LDS 1
#endif

#if defined(__HIP_DEVICE_COMPILE__) && __has_builtin(__builtin_amdgcn_s_wait_asynccnt)
#define WAIT_ASYNC(n) __builtin_amdgcn_s_wait_asynccnt(n)
#elif defined(__HIP_DEVICE_COMPILE__)
#define WAIT_ASYNC(n) asm volatile("s_wait_asynccnt %0" ::"i"(n) : "memory")
#else
#define WAIT_ASYNC(n)
#endif

__device__ __forceinline__ void async_cp_b128(const float* gsrc, float* ldst) {
#if defined(HAVE_ASYNC_LDS)
    // GLOBAL_LOAD_ASYNC_TO_LDS_B128: 16B per lane, tracked by ASYNCcnt,
    // bypasses the register file entirely. Param 0 is v4i addrspace(1)*
    // (per the round-2 diagnostic), param 1 the LDS destination.
    __builtin_amdgcn_global_load_async_to_lds_b128(
        (__attribute__((address_space(1))) v4i_vs*)(uintptr_t)gsrc,
        (__attribute__((address_space(3))) v4i_vs*)(uintptr_t)ldst,
        /*imm offset*/ 0, /*cpol*/ 0);
#else
    *(float4*)ldst = *(const float4*)gsrc;   // fallback: through VGPRs
#endif
}

__device__ __forceinline__ float seg_bilinear(const float* __restrict__ sp, int hh, int ww) {
    // Half-pixel 2x upsample: i0=(i>>1)+((i&1)?0:-1), frac=(i&1)?0.25:0.75.
    int h0 = (hh >> 1) + ((hh & 1) ? 0 : -1);
    int w0 = (ww >> 1) + ((ww & 1) ? 0 : -1);
    float fh = (hh & 1) ? 0.25f : 0.75f;
    float fw = (ww & 1) ? 0.25f : 0.75f;
    int h0c = h0 < 0 ? 0 : h0;
    int h1c = (h0 + 1) > (HS - 1) ? (HS - 1) : (h0 + 1);
    int w0c = w0 < 0 ? 0 : w0;
    int w1c = (w0 + 1) > (HS - 1) ? (HS - 1) : (w0 + 1);
    float v00 = sp[h0c * HS + w0c], v01 = sp[h0c * HS + w1c];
    float v10 = sp[h1c * HS + w0c], v11 = sp[h1c * HS + w1c];
    return (1.f - fh) * ((1.f - fw) * v00 + fw * v01)
         +        fh  * ((1.f - fw) * v10 + fw * v11);
}

__global__ __launch_bounds__(256) void zero_kernel(float* __restrict__ p, int n) {
    int i = blockIdx.x * blockDim.x + threadIdx.x;
    if (i < n) p[i] = 0.f;
}

// ---------------------------------------------------------------------------
// Pass 1: sums[n,k,c] = sum_hw fe*segup ; nump[n,k] = sum_hw segup
// Per-n GEMM [KP x HW]*[HW x C], HW = WMMA K dim. Double-buffered LDS tiles,
// async B128 staging of fe, VALU bilinear for seg rows, f32 atomics at end.
// ---------------------------------------------------------------------------
__device__ __forceinline__ void stats_stage(float (*dfe)[LPAD], float (*dseg)[LPAD],
                                            const float* __restrict__ feN,
                                            const float* __restrict__ segN,
                                            int p0, int tid) {
    // fe tile: 64 rows x 128 px = 2048 x 16B chunks / 256 thr = 8 async/thread
    for (int i = tid; i < CCH * 32; i += 256) {
        int cc = i >> 5, q = (i & 31) * 4;
        async_cp_b128(feN + (size_t)cc * HWPX + p0 + q, &dfe[cc][q]);
    }
    // seg rows 0..20: bilinear on the fly (source is 11 MB, L2-hot)
    for (int i = tid; i < KCL * 128; i += 256) {
        int kk = i >> 7, pp = i & 127;
        int p = p0 + pp;
        dseg[kk][pp] = seg_bilinear(segN + (size_t)kk * HS * HS, p >> 8, p & 255);
    }
}

__global__ __launch_bounds__(256) void stats_kernel(const float* __restrict__ fe,
                                                    const float* __restrict__ seg,
                                                    float* __restrict__ sums,   // [NB][KP][CCH]
                                                    float* __restrict__ nump) { // [NB][KP]
    __shared__ float lfe[2][CCH][LPAD];   // 2 x 33.8 KB
    __shared__ float lseg[2][KP][LPAD];   // 2 x 16.9 KB
    __shared__ float lnum[KCL];

    const int n    = blockIdx.y;
    const int tid  = threadIdx.x;
    const int lane = tid & 31;
    const int wave = tid >> 5;
    const int ktile = wave >> 2;
    const int ctile = wave & 3;
    const int m  = lane & 15;
    const int hi = lane >> 4;

    // zero padded seg rows 21..31 in both buffers (A rows must be 0 there)
    for (int i = tid; i < 2 * (KP - KCL) * 128; i += 256) {
        int bset = i / ((KP - KCL) * 128);
        int r    = (i / 128) % (KP - KCL);
        lseg[bset][KCL + r][i & 127] = 0.f;
    }
    if (tid < KCL) lnum[tid] = 0.f;

    const float* feN  = fe  + (size_t)n * CCH * HWPX;
    const float* segN = seg + (size_t)n * KCL * HS * HS;
    const int chunk0  = blockIdx.x * 1024;   // 64 chunks of 1024 px

    stats_stage(lfe[0], lseg[0], feN, segN, chunk0, tid);   // prologue

    v8f acc = {};
    float numacc = 0.f;

    for (int st = 0; st < 8; ++st) {
        const int cur = st & 1;
        if (st + 1 < 8) {
            stats_stage(lfe[cur ^ 1], lseg[cur ^ 1], feN, segN,
                        chunk0 + (st + 1) * 128, tid);
            WAIT_ASYNC(8);   // in-order retire: <=8 left => current tile landed
        } else {
            WAIT_ASYNC(0);
        }
        __syncthreads();

        if (tid < KCL * 8) {        // num_pixel partials
            int kk = tid >> 3, part = tid & 7;
            float s = 0.f;
            for (int q = 0; q < 16; ++q) s += lseg[cur][kk][part * 16 + q];
            numacc += s;
        }
        for (int p4 = 0; p4 < 128; p4 += 4) {
            v2f A, B;
            A.x = lseg[cur][ktile * 16 + m][p4 + 2 * hi + 0];  // A[M=k][K=px]
            A.y = lseg[cur][ktile * 16 + m][p4 + 2 * hi + 1];
            B.x = lfe[cur][ctile * 16 + m][p4 + 2 * hi + 0];   // B[K=px][N=c]
            B.y = lfe[cur][ctile * 16 + m][p4 + 2 * hi + 1];
            acc = __builtin_amdgcn_wmma_f32_16x16x4_f32(
                false, A, false, B, (short)0, acc, false, false);
        }
        __syncthreads();   // readers done before this buffer is re-staged
    }

    if (tid < KCL * 8) atomicAdd(&lnum[tid >> 3], numacc);
    __syncthreads();
    if (tid < KCL) atomicAdd(&nump[n * KP + tid], lnum[tid]);

    for (int v = 0; v < 8; ++v) {    // C/D: VGPR v -> M=v+8*hi, N=lane&15
        int kk = ktile * 16 + v + 8 * hi;
        int cc = ctile * 16 + m;
        if (kk < KCL) atomicAdd(&sums[((size_t)n * KP + kk) * CCH + cc], acc[v]);
    }
}

// ---------------------------------------------------------------------------
// Pass 2 (tiny): a = sums/nump ; W2[n] = [ w^T rows | (a@w^T) rows | 0 ]
// ---------------------------------------------------------------------------
__global__ __launch_bounds__(256) void finalize_kernel(const float* __restrict__ sums,
                                                       const float* __restrict__ nump,
                                                       const float* __restrict__ w,
                                                       float* __restrict__ W2) { // [NB][JTOT][CCH]
    __shared__ float la[KCL][CCH];
    __shared__ float lw[CCH][CCH];
    const int n = blockIdx.x, tid = threadIdx.x;

    for (int i = tid; i < CCH * CCH; i += 256) lw[i >> 6][i & 63] = w[i];
    for (int i = tid; i < KCL * CCH; i += 256) {
        int kk = i >> 6, cc = i & 63;
        la[kk][cc] = sums[((size_t)n * KP + kk) * CCH + cc] / nump[n * KP + kk];
    }
    __syncthreads();
    for (int i = tid; i < CCH * CCH; i += 256) {
        int j = i >> 6, o = i & 63;
        W2[((size_t)n * JTOT + j) * CCH + o] = lw[o][j];
    }
    for (int i = tid; i < KP * CCH; i += 256) {
        int kk = i >> 6, o = i & 63;
        float s = 0.f;
        if (kk < KCL) {
            for (int c = 0; c < CCH; ++c) s += la[kk][c] * lw[o][c];
        }
        W2[((size_t)n * JTOT + CCH + kk) * CCH + o] = s;
    }
}

// ---------------------------------------------------------------------------
// Pass 3: fused apply GEMM. y[o,px] = sum_j W2[j][o]*X[j][px] + b[o], LeakyReLU.
// X = [fe ; segup ; 0] double-buffered in LDS; 24 wmma (K=96) per 16x16 tile.
// ---------------------------------------------------------------------------
__device__ __forceinline__ void apply_stage(float (*dx)[LPAD],
                                            const float* __restrict__ feN,
                                            const float* __restrict__ segN,
                                            int p0, int tid) {
    for (int i = tid; i < CCH * 32; i += 256) {
        int cc = i >> 5, q = (i & 31) * 4;
        async_cp_b128(feN + (size_t)cc * HWPX + p0 + q, &dx[cc][q]);
    }
    for (int i = tid; i < KCL * 128; i += 256) {
        int kk = i >> 7, pp = i & 127;
        int p = p0 + pp;
        dx[CCH + kk][pp] = seg_bilinear(segN + (size_t)kk * HS * HS, p >> 8, p & 255);
    }
}

__global__ __launch_bounds__(256) void apply_kernel(const float* __restrict__ fe,
                                                    const float* __restrict__ seg,
                                                    const float* __restrict__ W2,
                                                    const float* __restrict__ b,
                                                    float* __restrict__ out) {
    __shared__ float lx[2][JTOT][LPAD];   // 2 x 50.7 KB
    __shared__ float lw2[JTOT][CCH];      // 24.6 KB
    __shared__ float lb[CCH];

    const int n    = blockIdx.y;
    const int tid  = threadIdx.x;
    const int lane = tid & 31;
    const int wave = tid >> 5;            // pixel tile 0..7
    const int m  = lane & 15;
    const int hi = lane >> 4;

    for (int i = tid; i < JTOT * CCH; i += 256)
        lw2[i >> 6][i & 63] = W2[(size_t)n * JTOT * CCH + i];
    if (tid < CCH) lb[tid] = b[tid];
    for (int i = tid; i < 2 * (JTOT - (CCH + KCL)) * 128; i += 256) {   // zero pad rows
        int bset = i / ((JTOT - (CCH + KCL)) * 128);
        int r    = (i / 128) % (JTOT - (CCH + KCL));
        lx[bset][CCH + KCL + r][i & 127] = 0.f;
    }

    const float* feN  = fe  + (size_t)n * CCH * HWPX;
    const float* segN = seg + (size_t)n * KCL * HS * HS;
    const int chunk0  = blockIdx.x * 2048;   // 32 chunks of 2048 px

    apply_stage(lx[0], feN, segN, chunk0, tid);   // prologue

    for (int st = 0; st < 16; ++st) {
        const int cur = st & 1;
        const int p0  = chunk0 + st * 128;
        if (st + 1 < 16) {
            apply_stage(lx[cur ^ 1], feN, segN, chunk0 + (st + 1) * 128, tid);
            WAIT_ASYNC(8);
        } else {
            WAIT_ASYNC(0);
        }
        __syncthreads();

        for (int ot = 0; ot < 4; ++ot) {         // 4 o-tiles of 16
            v8f acc = {};
            for (int j0 = 0; j0 < JTOT; j0 += 4) {   // K=96 -> 24 wmma
                v2f A, B;
                A.x = lw2[j0 + 2 * hi + 0][ot * 16 + m];   // A[M=o][K=j]
                A.y = lw2[j0 + 2 * hi + 1][ot * 16 + m];
                B.x = lx[cur][j0 + 2 * hi + 0][wave * 16 + m];  // B[K=j][N=px]
                B.y = lx[cur][j0 + 2 * hi + 1][wave * 16 + m];
                acc = __builtin_amdgcn_wmma_f32_16x16x4_f32(
                    false, A, false, B, (short)0, acc, false, false);
            }
            for (int v = 0; v < 8; ++v) {
                int oo = ot * 16 + v + 8 * hi;
                float y = acc[v] + lb[oo];
                y = (y > 0.f) ? y : 0.2f * y;
                out[((size_t)n * CCH + oo) * HWPX + p0 + wave * 16 + m] = y;
            }
        }
        __syncthreads();
    }
}

// ---------------------------------------------------------------------------
extern "C" void kernel_launch(void* const* d_in, const int* in_sizes, int n_in,
                              void* d_out, int out_size, void* d_ws, size_t ws_size,
                              hipStream_t stream) {
    const float* fe  = (const float*)d_in[0];
    const float* seg = (const float*)d_in[1];
    const float* w   = (const float*)d_in[2];
    const float* b   = (const float*)d_in[3];
    float* out = (float*)d_out;

    float* ws   = (float*)d_ws;
    float* sums = ws;                          // NB*KP*CCH
    float* nump = sums + NB * KP * CCH;        // NB*KP
    float* W2   = nump + NB * KP;              // NB*JTOT*CCH

    const int zn = NB * KP * CCH + NB * KP;
    zero_kernel<<<(zn + 255) / 256, 256, 0, stream>>>(ws, zn);
    stats_kernel<<<dim3(HWPX / 1024, NB), 256, 0, stream>>>(fe, seg, sums, nump);
    finalize_kernel<<<dim3(NB), 256, 0, stream>>>(sums, nump, w, W2);
    apply_kernel<<<dim3(HWPX / 2048, NB), 256, 0, stream>>>(fe, seg, W2, b, out);
}